// BaseTransform_17549236372294
// MI455X (gfx1250) — compile-verified
//
#include <hip/hip_runtime.h>
#include <hip/hip_bf16.h>

#define IMG_H 256
#define IMG_W 704
#define FHH 32
#define FWW 88
#define DD  118
#define CCH 80
#define NCAM 6
#define NXV 360
#define NYV 360

#define CAMSTRIDE 56        // Minv[16] idainv[16] mat[16] n3[3] nP0 nP1 (53, padded)
#define PTS_PER_BLOCK 128
#define THREADS 256

// ---------- CDNA5 async global->LDS staging (guarded) ----------
#if defined(__HIP_DEVICE_COMPILE__) && defined(__gfx1250__)
#if __has_builtin(__builtin_amdgcn_global_load_async_to_lds_b128)
#define HAVE_ASYNC_LDS 1
#endif
#endif

typedef int v4i_ __attribute__((vector_size(16)));
#if defined(__HIP_DEVICE_COMPILE__)
typedef __attribute__((address_space(1))) v4i_* g_v4i_ptr;
typedef __attribute__((address_space(3))) v4i_* l_v4i_ptr;
#endif

__device__ __forceinline__ void stage16(const float* g, float* l) {
#ifdef HAVE_ASYNC_LDS
  __builtin_amdgcn_global_load_async_to_lds_b128(
      (g_v4i_ptr)(g),
      (l_v4i_ptr)(l),
      0, 0);
#else
  *(float4*)l = *(const float4*)g;
#endif
}

__device__ __forceinline__ void async_wait_all() {
#ifdef HAVE_ASYNC_LDS
#if __has_builtin(__builtin_amdgcn_s_wait_asynccnt)
  __builtin_amdgcn_s_wait_asynccnt(0);
#else
  asm volatile("s_wait_asynccnt 0" ::: "memory");
#endif
#endif
}

// ---------- kernel: fused geometry + async-staged scatter (HOT) ----------
__global__ __launch_bounds__(THREADS)
void bevpool_scatter(const float* __restrict__ x,
                     const float* __restrict__ cam,
                     float* __restrict__ out) {
  __shared__ float feat[PTS_PER_BLOCK * CCH];   // 40 KB
  __shared__ int   gx_s[PTS_PER_BLOCK];
  __shared__ int   gy_s[PTS_PER_BLOCK];

  const unsigned tid = threadIdx.x;
  const unsigned p0  = blockIdx.x * PTS_PER_BLOCK;

  // Phase A: async DMA of the block's contiguous feature tile into LDS.
  const float* gsrc = x + (size_t)p0 * CCH;
  #pragma unroll
  for (int i = 0; i < (PTS_PER_BLOCK * CCH) / (THREADS * 4); ++i) {
    const unsigned e4 = tid + (unsigned)i * THREADS;     // float4 index
    stage16(gsrc + (size_t)e4 * 4, &feat[e4 * 4]);
  }

  // Phase B: geometry for this block's 128 points (one camera per block:
  // 118*32*88 = 332288 is divisible by 128).
  if (tid < PTS_PER_BLOCK) {
    const unsigned p    = p0 + tid;
    const unsigned ncam = p / (DD * FHH * FWW);
    unsigned rem        = p - ncam * (DD * FHH * FWW);
    const unsigned di   = rem / (FHH * FWW);
    rem                -= di * (FHH * FWW);
    const unsigned h    = rem / FWW;
    const unsigned w    = rem - h * FWW;

    const float* c  = cam + ncam * CAMSTRIDE;
    const float* Mi = c;          // inv(intr) @ inv(ida)
    const float* Ai = c + 16;     // inv(ida)
    const float* T  = c + 32;     // bda @ s2e @ inv(intr)

    const float fx = (float)w * ((float)(IMG_W - 1) / (float)(FWW - 1));
    const float fy = (float)h * ((float)(IMG_H - 1) / (float)(FHH - 1));
    const float t  = (float)di / (float)DD;
    const float dv = t * sqrtf(t);                       // (di/D)^1.5

    // ray dir in sensor frame
    const float rx = Mi[0]*fx + Mi[1]*fy + Mi[2]  + Mi[3];
    const float ry = Mi[4]*fx + Mi[5]*fy + Mi[6]  + Mi[7];
    const float rz = Mi[8]*fx + Mi[9]*fy + Mi[10] + Mi[11];
    const float rl = rsqrtf(rx*rx + ry*ry + rz*rz);
    const float ux = rx*rl, uy = ry*rl, uz = rz*rl;

    const float ndir  = c[48]*ux + c[49]*uy + c[50]*uz;
    const float t0    = c[51] / ndir;
    const float tdiff = t0 - c[52] / ndir;
    const float z     = (t0 - dv * tdiff) * uz;

    // q = inv(ida) @ (fx, fy, z, 1)
    const float qx = Ai[0]*fx  + Ai[1]*fy  + Ai[2]*z  + Ai[3];
    const float qy = Ai[4]*fx  + Ai[5]*fy  + Ai[6]*z  + Ai[7];
    const float qz = Ai[8]*fx  + Ai[9]*fy  + Ai[10]*z + Ai[11];
    const float qw = Ai[12]*fx + Ai[13]*fy + Ai[14]*z + Ai[15];
    const float px = qx * qz, py = qy * qz, pz = qz, pw = qw;

    const float gxf = (T[0]*px + T[1]*py + T[2]*pz  + T[3]*pw  + 54.0f) / 0.3f;
    const float gyf = (T[4]*px + T[5]*py + T[6]*pz  + T[7]*pw  + 54.0f) / 0.3f;
    const float gzf = (T[8]*px + T[9]*py + T[10]*pz + T[11]*pw + 10.0f) / 20.0f;

    const int gx = (int)gxf;   // C truncation == numpy astype(int32)
    const int gy = (int)gyf;
    const int gz = (int)gzf;
    const bool kept = (gx >= 0) && (gx < NXV) && (gy >= 0) && (gy < NYV) &&
                      (gz >= 0) && (gz < 1);
    gx_s[tid] = kept ? gx : -1;
    gy_s[tid] = gy;
  }

  async_wait_all();       // ASYNCcnt == 0 for this wave's DMAs
  __syncthreads();        // all waves' DMAs + geometry visible

  // Phase C: scatter 128*80 (point,channel) pairs, 40 per thread.
  unsigned j  = tid;              // linear LDS index == pl*80 + ch
  unsigned pl = tid / CCH;
  unsigned ch = tid - pl * CCH;
  #pragma unroll 4
  for (int it = 0; it < (PTS_PER_BLOCK * CCH) / THREADS; ++it) {
    const int gx = gx_s[pl];
    if (gx >= 0) {
      const int gy = gy_s[pl];
      float* dst = out + ((size_t)ch * NXV + (unsigned)gx) * NYV + (unsigned)gy;
      (void)__hip_atomic_fetch_add(dst, feat[j], __ATOMIC_RELAXED,
                                   __HIP_MEMORY_SCOPE_AGENT);
    }
    j += THREADS;
    pl += 3; ch += 16;                 // 256 = 3*80 + 16
    if (ch >= CCH) { ch -= CCH; ++pl; }
  }
}

// ---------- small matrix helpers ----------
__device__ __forceinline__ void mat4_mul(const float* A, const float* B, float* C) {
  for (int i = 0; i < 4; ++i)
    for (int j = 0; j < 4; ++j) {
      float s = 0.f;
      for (int k = 0; k < 4; ++k) s += A[i*4+k] * B[k*4+j];
      C[i*4+j] = s;
    }
}

// Adjugate-based 4x4 inverse (works identically for row- or column-major).
__device__ void mat4_inv(const float* m, float* o) {
  float inv[16];
  inv[0]  =  m[5]*m[10]*m[15] - m[5]*m[11]*m[14] - m[9]*m[6]*m[15] + m[9]*m[7]*m[14] + m[13]*m[6]*m[11] - m[13]*m[7]*m[10];
  inv[4]  = -m[4]*m[10]*m[15] + m[4]*m[11]*m[14] + m[8]*m[6]*m[15] - m[8]*m[7]*m[14] - m[12]*m[6]*m[11] + m[12]*m[7]*m[10];
  inv[8]  =  m[4]*m[9]*m[15]  - m[4]*m[11]*m[13] - m[8]*m[5]*m[15] + m[8]*m[7]*m[13] + m[12]*m[5]*m[11] - m[12]*m[7]*m[9];
  inv[12] = -m[4]*m[9]*m[14]  + m[4]*m[10]*m[13] + m[8]*m[5]*m[14] - m[8]*m[6]*m[13] - m[12]*m[5]*m[10] + m[12]*m[6]*m[9];
  inv[1]  = -m[1]*m[10]*m[15] + m[1]*m[11]*m[14] + m[9]*m[2]*m[15] - m[9]*m[3]*m[14] - m[13]*m[2]*m[11] + m[13]*m[3]*m[10];
  inv[5]  =  m[0]*m[10]*m[15] - m[0]*m[11]*m[14] - m[8]*m[2]*m[15] + m[8]*m[3]*m[14] + m[12]*m[2]*m[11] - m[12]*m[3]*m[10];
  inv[9]  = -m[0]*m[9]*m[15]  + m[0]*m[11]*m[13] + m[8]*m[1]*m[15] - m[8]*m[3]*m[13] - m[12]*m[1]*m[11] + m[12]*m[3]*m[9];
  inv[13] =  m[0]*m[9]*m[14]  - m[0]*m[10]*m[13] - m[8]*m[1]*m[14] + m[8]*m[2]*m[13] + m[12]*m[1]*m[10] - m[12]*m[2]*m[9];
  inv[2]  =  m[1]*m[6]*m[15]  - m[1]*m[7]*m[14]  - m[5]*m[2]*m[15] + m[5]*m[3]*m[14] + m[13]*m[2]*m[7]  - m[13]*m[3]*m[6];
  inv[6]  = -m[0]*m[6]*m[15]  + m[0]*m[7]*m[14]  + m[4]*m[2]*m[15] - m[4]*m[3]*m[14] - m[12]*m[2]*m[7]  + m[12]*m[3]*m[6];
  inv[10] =  m[0]*m[5]*m[15]  - m[0]*m[7]*m[13]  - m[4]*m[1]*m[15] + m[4]*m[3]*m[13] + m[12]*m[1]*m[7]  - m[12]*m[3]*m[5];
  inv[14] = -m[0]*m[5]*m[14]  + m[0]*m[6]*m[13]  + m[4]*m[1]*m[14] - m[4]*m[2]*m[13] - m[12]*m[1]*m[6]  + m[12]*m[2]*m[5];
  inv[3]  = -m[1]*m[6]*m[11]  + m[1]*m[7]*m[10]  + m[5]*m[2]*m[11] - m[5]*m[3]*m[10] - m[9]*m[2]*m[7]   + m[9]*m[3]*m[6];
  inv[7]  =  m[0]*m[6]*m[11]  - m[0]*m[7]*m[10]  - m[4]*m[2]*m[11] + m[4]*m[3]*m[10] + m[8]*m[2]*m[7]   - m[8]*m[3]*m[6];
  inv[11] = -m[0]*m[5]*m[11]  + m[0]*m[7]*m[9]   + m[4]*m[1]*m[11] - m[4]*m[3]*m[9]  - m[8]*m[1]*m[7]   + m[8]*m[3]*m[5];
  inv[15] =  m[0]*m[5]*m[10]  - m[0]*m[6]*m[9]   - m[4]*m[1]*m[10] + m[4]*m[2]*m[9]  + m[8]*m[1]*m[6]   - m[8]*m[2]*m[5];
  float det = m[0]*inv[0] + m[1]*inv[4] + m[2]*inv[8] + m[3]*inv[12];
  det = 1.0f / det;
  for (int i = 0; i < 16; ++i) o[i] = inv[i] * det;
}

// ---------- kernel: per-camera constants ----------
__global__ void precompute_cams(const float* __restrict__ c2l,
                                const float* __restrict__ intr,
                                const float* __restrict__ ida,
                                const float* __restrict__ bda,
                                const float* __restrict__ den,
                                float* __restrict__ cam) {
  const int n = threadIdx.x;
  if (n >= NCAM) return;
  float M[16], Mi[16], In[16], Ii[16], Ia[16], Iai[16], Bd[16];
  for (int i = 0; i < 16; ++i) {
    M[i]  = c2l[n*16 + i];
    In[i] = intr[n*16 + i];
    Ia[i] = ida[n*16 + i];
    Bd[i] = bda[i];
  }
  mat4_inv(M, Mi);
  mat4_inv(In, Ii);
  mat4_inv(Ia, Iai);
  const float ox = Mi[3], oy = Mi[7], oz = Mi[11];          // ego2sensor[:3,3]
  float nx = den[n*4+0], ny = den[n*4+1], nz = den[n*4+2];
  const float il = rsqrtf(nx*nx + ny*ny + nz*nz);
  nx *= il; ny *= il; nz *= il;
  const float nP0 = nx*(ox + 1.0f*nx)  + ny*(oy + 1.0f*ny)  + nz*(oz + 1.0f*nz);
  const float nP1 = nx*(ox + 60.0f*nx) + ny*(oy + 60.0f*ny) + nz*(oz + 60.0f*nz);
  float Minv[16], T1[16], T2[16];
  mat4_mul(Ii, Iai, Minv);            // inv(intr) @ inv(ida)
  mat4_mul(M, Ii, T1);                // sensor2ego @ inv(intr)
  mat4_mul(Bd, T1, T2);               // bda @ (...)
  float* o = cam + n*CAMSTRIDE;
  for (int i = 0; i < 16; ++i) o[i]      = Minv[i];
  for (int i = 0; i < 16; ++i) o[16 + i] = Iai[i];
  for (int i = 0; i < 16; ++i) o[32 + i] = T2[i];
  o[48] = nx; o[49] = ny; o[50] = nz; o[51] = nP0; o[52] = nP1;
}

// ---------- kernel: zero the accumulator ----------
__global__ void zero_out(float4* __restrict__ o4, int n4) {
  const int i = blockIdx.x * blockDim.x + threadIdx.x;
  if (i < n4) o4[i] = make_float4(0.f, 0.f, 0.f, 0.f);
}

extern "C" void kernel_launch(void* const* d_in, const int* in_sizes, int n_in,
                              void* d_out, int out_size, void* d_ws, size_t ws_size,
                              hipStream_t stream) {
  const float* x    = (const float*)d_in[0];
  const float* c2l  = (const float*)d_in[1];
  const float* intr = (const float*)d_in[2];
  const float* ida  = (const float*)d_in[3];
  const float* bda  = (const float*)d_in[4];
  const float* den  = (const float*)d_in[5];
  float* out = (float*)d_out;
  float* cam = (float*)d_ws;                 // 6*56*4 = 1344 bytes

  precompute_cams<<<1, 32, 0, stream>>>(c2l, intr, ida, bda, den, cam);

  const int n4 = out_size / 4;               // out_size = 10,368,000
  zero_out<<<(n4 + 255) / 256, 256, 0, stream>>>((float4*)out, n4);

  const unsigned Np = NCAM * DD * FHH * FWW; // 1,993,728
  bevpool_scatter<<<Np / PTS_PER_BLOCK, THREADS, 0, stream>>>(x, cam, out);
}